// QuantizeAdaptive_31353261261011
// MI455X (gfx1250) — compile-verified
//
#include <hip/hip_runtime.h>

typedef __attribute__((ext_vector_type(2))) float v2f;
typedef __attribute__((ext_vector_type(8))) float v8f;

#define DECAY 0.99f
#define EPSV  1e-5f
#define NROWS 131072   // 32*64*64
#define DIM   64
#define NCODE 512
#define XTOT  8388608  // NROWS*DIM
#define TILES 4        // 16-row tiles per wave -> 64 rows/wave

// d_out layout (floats, concatenated in reference return order):
//  out0 quantize          : 8388608
//  out1 diff              : 1
//  out2 embed_ind (float) : 131072
//  out3 new_embed         : 32768
//  out4 new_cluster_size  : 512   (used as atomic count accumulator first)
//  out5 new_embed_avg     : 32768 (used as atomic embed_sum accumulator first)

__global__ __launch_bounds__(256) void vq_init(const float* __restrict__ embed,
                                               float* __restrict__ out4,
                                               float* __restrict__ out5,
                                               float* __restrict__ esq,
                                               float* __restrict__ diff_acc) {
  int i = blockIdx.x * blockDim.x + threadIdx.x;
  if (i < 32768) out5[i] = 0.0f;
  if (i < NCODE) {
    out4[i] = 0.0f;
    float s = 0.0f;
#pragma unroll 8
    for (int d = 0; d < DIM; ++d) {
      float v = embed[d * NCODE + i];
      s += v * v;
    }
    esq[i] = s;
  }
  if (i == 0) diff_acc[0] = 0.0f;
}

__global__ __launch_bounds__(256) void vq_main(const float* __restrict__ x,
                                               const float* __restrict__ embed,
                                               const float* __restrict__ esq,
                                               float* __restrict__ quant,
                                               float* __restrict__ eind,
                                               float* __restrict__ counts,
                                               float* __restrict__ esum,
                                               float* __restrict__ diff_acc) {
  const int lane = threadIdx.x & 31;
  const int wave = threadIdx.x >> 5;
  const int wtile = blockIdx.x * 8 + wave;  // 256 blocks * 8 waves = 2048
  const int row0 = wtile * (16 * TILES);    // 64 rows per wave
  const int j = lane & 15;                  // N-lane / M-row selector
  const int h = lane >> 4;                  // half of wave -> K pair / M offset

  // ---- preload A tiles (TILES x 16 rows x 64 dims), reused for all 32 chunks ----
  // A layout (16x4 f32): lane L: M = L%16; VGPR0 = A[M][2h+4t], VGPR1 = A[M][2h+4t+1]
  v2f a[TILES][16];
#pragma unroll
  for (int tt = 0; tt < TILES; ++tt) {
    const float* xr = x + (size_t)(row0 + 16 * tt + j) * DIM + 2 * h;
#pragma unroll
    for (int t = 0; t < 16; ++t)
      a[tt][t] = *(const v2f*)(xr + 4 * t);
  }

  float best[TILES][8];
  int bidx[TILES][8];
#pragma unroll
  for (int tt = 0; tt < TILES; ++tt)
#pragma unroll
    for (int v = 0; v < 8; ++v) { best[tt][v] = 3.4e38f; bidx[tt][v] = 0x7fffffff; }

  // ---- loop over 32 chunks of 16 codes; B loaded once, shared by TILES accs ----
  for (int c = 0; c < 32; ++c) {
    const int n0 = c * 16;
    const float* ec = embed + n0 + j;   // embed[k][n] = embed[k*512 + n]
    v8f acc[TILES];
#pragma unroll
    for (int tt = 0; tt < TILES; ++tt) acc[tt] = (v8f){};

    // 2-deep software pipeline on B: issue loads for step t+1 before WMMAs of t
    v2f bc;
    {
      const int k0 = 2 * h;
      bc.x = ec[(size_t)k0 * NCODE];
      bc.y = ec[(size_t)(k0 + 1) * NCODE];
    }
#pragma unroll
    for (int t = 0; t < 16; ++t) {
      v2f bn;
      if (t < 15) {
        const int k = 4 * (t + 1) + 2 * h;
        bn.x = ec[(size_t)k * NCODE];
        bn.y = ec[(size_t)(k + 1) * NCODE];
      }
#pragma unroll
      for (int tt = 0; tt < TILES; ++tt)
        acc[tt] = __builtin_amdgcn_wmma_f32_16x16x4_f32(false, a[tt][t], false, bc,
                                                        (short)0, acc[tt], false, false);
      if (t < 15) bc = bn;
    }

    // lane holds C[v] for (M = v + 8h, N = n0 + j); argmin key = |e|^2 - 2*x.e
    const float eq = esq[n0 + j];
    const int myn = n0 + j;
#pragma unroll
    for (int tt = 0; tt < TILES; ++tt) {
#pragma unroll
      for (int v = 0; v < 8; ++v) {
        float key = eq - 2.0f * acc[tt][v];
        if (key < best[tt][v] || (key == best[tt][v] && myn < bidx[tt][v])) {
          best[tt][v] = key;
          bidx[tt][v] = myn;
        }
      }
    }
  }

  // ---- argmin reduction across the 16 lanes of each half (masks stay in-half) ----
#pragma unroll
  for (int tt = 0; tt < TILES; ++tt) {
#pragma unroll
    for (int v = 0; v < 8; ++v) {
#pragma unroll
      for (int m = 1; m <= 8; m <<= 1) {
        float ov = __shfl_xor(best[tt][v], m, 32);
        int oi = __shfl_xor(bidx[tt][v], m, 32);
        if (ov < best[tt][v] || (ov == best[tt][v] && oi < bidx[tt][v])) {
          best[tt][v] = ov;
          bidx[tt][v] = oi;
        }
      }
    }
  }

  if (j == 0) {
#pragma unroll
    for (int tt = 0; tt < TILES; ++tt)
#pragma unroll
      for (int v = 0; v < 8; ++v)
        eind[row0 + 16 * tt + 8 * h + v] = (float)bidx[tt][v];
  }

  // ---- quantize gather + diff + EMA segment sums ----
  float dacc = 0.0f;
#pragma unroll
  for (int tt = 0; tt < TILES; ++tt) {
#pragma unroll
    for (int v = 0; v < 8; ++v) {
      const int row = row0 + 16 * tt + 8 * h + v;
      const int idx = bidx[tt][v];
      const int d0 = 4 * j;               // 16 lanes * 4 dims = 64 dims per row
      const float* eqp = embed + idx;
      float q0 = eqp[(size_t)(d0 + 0) * NCODE];
      float q1 = eqp[(size_t)(d0 + 1) * NCODE];
      float q2 = eqp[(size_t)(d0 + 2) * NCODE];
      float q3 = eqp[(size_t)(d0 + 3) * NCODE];
      const float4 xv = *(const float4*)(x + (size_t)row * DIM + d0);
      float4 qv = make_float4(q0, q1, q2, q3);
      *(float4*)(quant + (size_t)row * DIM + d0) = qv;
      float e0 = q0 - xv.x, e1 = q1 - xv.y, e2 = q2 - xv.z, e3 = q3 - xv.w;
      dacc += e0 * e0 + e1 * e1 + e2 * e2 + e3 * e3;
      atomicAdd(&esum[(size_t)(d0 + 0) * NCODE + idx], xv.x);
      atomicAdd(&esum[(size_t)(d0 + 1) * NCODE + idx], xv.y);
      atomicAdd(&esum[(size_t)(d0 + 2) * NCODE + idx], xv.z);
      atomicAdd(&esum[(size_t)(d0 + 3) * NCODE + idx], xv.w);
      if (j == 0) atomicAdd(&counts[idx], 1.0f);
    }
  }

#pragma unroll
  for (int m = 16; m >= 1; m >>= 1)
    dacc += __shfl_xor(dacc, m, 32);
  if (lane == 0) atomicAdd(diff_acc, dacc);
}

__global__ __launch_bounds__(512) void vq_fin(const float* __restrict__ cluster_size,
                                              const float* __restrict__ embed_avg,
                                              float* __restrict__ out1,
                                              float* __restrict__ out3,
                                              float* __restrict__ out4,
                                              float* __restrict__ out5,
                                              const float* __restrict__ diff_acc) {
  __shared__ float red[512];
  const int t = threadIdx.x;
  float ncs = cluster_size[t] * DECAY + (1.0f - DECAY) * out4[t];
  out4[t] = ncs;
  red[t] = ncs;
  __syncthreads();
  for (int s = 256; s > 0; s >>= 1) {
    if (t < s) red[t] += red[t + s];
    __syncthreads();
  }
  const float n = red[0];
  const float cs = (ncs + EPSV) / (n + (float)NCODE * EPSV) * n;
  for (int d = 0; d < DIM; ++d) {
    int i = d * NCODE + t;
    float nea = embed_avg[i] * DECAY + (1.0f - DECAY) * out5[i];
    out5[i] = nea;
    out3[i] = nea / cs;
  }
  if (t == 0) out1[0] = diff_acc[0] / (float)XTOT;
}

extern "C" void kernel_launch(void* const* d_in, const int* in_sizes, int n_in,
                              void* d_out, int out_size, void* d_ws, size_t ws_size,
                              hipStream_t stream) {
  const float* x            = (const float*)d_in[0];
  const float* embed        = (const float*)d_in[1];
  const float* cluster_size = (const float*)d_in[2];
  const float* embed_avg    = (const float*)d_in[3];

  float* out0 = (float*)d_out;           // quantize
  float* out1 = out0 + XTOT;             // diff
  float* out2 = out1 + 1;                // embed_ind
  float* out3 = out2 + NROWS;            // new_embed
  float* out4 = out3 + DIM * NCODE;      // new_cluster_size
  float* out5 = out4 + NCODE;            // new_embed_avg

  float* esq      = (float*)d_ws;        // 512 floats
  float* diff_acc = esq + NCODE;         // 1 float

  vq_init<<<128, 256, 0, stream>>>(embed, out4, out5, esq, diff_acc);
  vq_main<<<256, 256, 0, stream>>>(x, embed, esq, out0, out2, out4, out5, diff_acc);
  vq_fin<<<1, 512, 0, stream>>>(cluster_size, embed_avg, out1, out3, out4, out5, diff_acc);
}